// GNN_5377299055106
// MI455X (gfx1250) — compile-verified
//
#include <hip/hip_runtime.h>
#include <math.h>
#include <stdint.h>

// Problem constants (match reference)
#define NN   40000   // nodes
#define EE   640000  // edges
#define TOPF 12      // input features
#define HIDF 128     // hidden width

typedef __attribute__((ext_vector_type(2))) float v2f;
typedef __attribute__((ext_vector_type(8))) float v8f;

// ---------------- degree / inverse degree ----------------
__global__ __launch_bounds__(256)
void deg_kernel(const int* __restrict__ dst, float* __restrict__ deg) {
    int e = blockIdx.x * blockDim.x + threadIdx.x;
    if (e < EE) atomicAdd(&deg[dst[e]], 1.0f);
}

__global__ __launch_bounds__(256)
void invdeg_kernel(float* __restrict__ deg) {
    int n = blockIdx.x * blockDim.x + threadIdx.x;
    if (n < NN) deg[n] = 1.0f / fmaxf(deg[n], 1.0f);
}

// ---------------- scatter-add aggregation (mean numerator) ----------------
// One thread per (edge, 4-feature group): float4 gather + 4 scalar atomics.
// A wave covers one edge's full feature row (F=128) -> coalesced 512B gather.
template<int F>
__global__ __launch_bounds__(256)
void scatter_kernel(const float* __restrict__ h, const int* __restrict__ src,
                    const int* __restrict__ dst, float* __restrict__ agg) {
    constexpr int G = F / 4;
    int t = blockIdx.x * blockDim.x + threadIdx.x;
    if (t >= EE * G) return;
    int e = t / G;
    int g = (t - e * G) * 4;
    int s = src[e];
    int d = dst[e];
    const float4 v = *(const float4*)(h + (long long)s * F + g);
    float* a = agg + (long long)d * F + g;
    atomicAdd(a + 0, v.x);
    atomicAdd(a + 1, v.y);
    atomicAdd(a + 2, v.z);
    atomicAdd(a + 3, v.w);
}

// ---------------- fused SAGE layer GEMM (WMMA f32 16x16x4) ----------------
// out[n, f] = elu( sum_k agg[n,k]*invdeg[n]*Wl[f,k] + bl[f] + sum_k hin[n,k]*Wr[f,k] )
// Block = 256 threads = 8 waves. Weight tiles for this block's 16 output
// columns (one contiguous 16xK slab each of Wl and Wr) are staged into LDS
// once per block via async global->LDS copies (ASYNCcnt). Each wave then
// computes a 32x16 output tile (two accumulators sharing one B fragment),
// so a block covers 256 rows. Grid: (ceil(NN/256), HIDF/16).
template<int K, bool ELU>
__global__ __launch_bounds__(256)
void sage_gemm_kernel(const float* __restrict__ agg, const float* __restrict__ invdeg,
                      const float* __restrict__ hin,
                      const float* __restrict__ Wl, const float* __restrict__ bl,
                      const float* __restrict__ Wr, float* __restrict__ out) {
    __shared__ float sB[2][16][K];   // [0]=Wl tile, [1]=Wr tile (row n = tile col)

    const int tid  = threadIdx.x;
    const int col0 = blockIdx.y * 16;

    // ---- async stage: Wl/Wr rows [col0, col0+16) -> LDS (contiguous 16*K floats each)
    {
        const unsigned lbase0 = (unsigned)(uintptr_t)&sB[0][0][0];
        const unsigned lbase1 = (unsigned)(uintptr_t)&sB[1][0][0];
        const float* g0 = Wl + (size_t)col0 * K;
        const float* g1 = Wr + (size_t)col0 * K;
        for (int i = tid; i < (16 * K) / 2; i += 256) {
            const unsigned l0 = lbase0 + i * 8u;
            const unsigned l1 = lbase1 + i * 8u;
            const float* p0 = g0 + i * 2;
            const float* p1 = g1 + i * 2;
            asm volatile("global_load_async_to_lds_b64 %0, %1, off"
                         :: "v"(l0), "v"(p0) : "memory");
            asm volatile("global_load_async_to_lds_b64 %0, %1, off"
                         :: "v"(l1), "v"(p1) : "memory");
        }
        asm volatile("s_wait_asynccnt 0x0" ::: "memory");
        __syncthreads();
    }

    const int lane = tid & 31;
    const int wave = tid >> 5;
    const int row0 = blockIdx.x * 256 + wave * 32;  // 32-row tile base for this wave

    const int m  = lane & 15;          // A row / B col / D col within tile
    const int kh = (lane >> 4) << 1;   // 0 or 2: K sub-offset for this half-wave

    int rowA0 = row0 + m;       if (rowA0 > NN - 1) rowA0 = NN - 1;
    int rowA1 = row0 + 16 + m;  if (rowA1 > NN - 1) rowA1 = NN - 1;

    const float  invd0 = invdeg[rowA0];
    const float  invd1 = invdeg[rowA1];
    const float* aRow0 = agg + (long long)rowA0 * K;
    const float* aRow1 = agg + (long long)rowA1 * K;
    const float* hRow0 = hin + (long long)rowA0 * K;
    const float* hRow1 = hin + (long long)rowA1 * K;

    v8f c0 = {};
    v8f c1 = {};

    #pragma unroll 4
    for (int k0 = 0; k0 < K; k0 += 4) {
        const int ka = k0 + kh;
        // B fragments from LDS: b[r] = W[col0+n][ka+r], n = lane&15
        const v2f b1 = *(const v2f*)&sB[0][m][ka];
        const v2f b2 = *(const v2f*)&sB[1][m][ka];

        // row group 0
        v2f a0 = *(const v2f*)(aRow0 + ka);
        a0.x *= invd0; a0.y *= invd0;
        c0 = __builtin_amdgcn_wmma_f32_16x16x4_f32(false, a0, false, b1,
                                                   (short)0, c0, false, false);
        const v2f h0 = *(const v2f*)(hRow0 + ka);
        c0 = __builtin_amdgcn_wmma_f32_16x16x4_f32(false, h0, false, b2,
                                                   (short)0, c0, false, false);
        // row group 1 (reuses b1/b2)
        v2f a1 = *(const v2f*)(aRow1 + ka);
        a1.x *= invd1; a1.y *= invd1;
        c1 = __builtin_amdgcn_wmma_f32_16x16x4_f32(false, a1, false, b1,
                                                   (short)0, c1, false, false);
        const v2f h1 = *(const v2f*)(hRow1 + ka);
        c1 = __builtin_amdgcn_wmma_f32_16x16x4_f32(false, h1, false, b2,
                                                   (short)0, c1, false, false);
    }

    // D layout: lane holds column n = lane&15; VGPR r holds row (r + 8*(lane>>4))
    const float bias  = bl[col0 + m];
    const int   half8 = (lane >> 4) * 8;
    #pragma unroll
    for (int r = 0; r < 8; ++r) {
        const int rowD0 = row0 + r + half8;
        if (rowD0 < NN) {
            float v = c0[r] + bias;
            if (ELU) v = (v > 0.0f) ? v : (__expf(v) - 1.0f);
            out[(long long)rowD0 * HIDF + col0 + m] = v;
        }
        const int rowD1 = row0 + 16 + r + half8;
        if (rowD1 < NN) {
            float v = c1[r] + bias;
            if (ELU) v = (v > 0.0f) ? v : (__expf(v) - 1.0f);
            out[(long long)rowD1 * HIDF + col0 + m] = v;
        }
    }
}

// ---------------- output head: h @ Wout.T + bout, one wave per node ----------------
__global__ __launch_bounds__(256)
void head_kernel(const float* __restrict__ h, const float* __restrict__ Wout,
                 const float* __restrict__ bout, float* __restrict__ out) {
    const int lane = threadIdx.x & 31;
    const int node = (blockIdx.x * blockDim.x + threadIdx.x) >> 5;
    if (node >= NN) return;
    const float4 a = ((const float4*)(h + (long long)node * HIDF))[lane];
    const float4 w = ((const float4*)Wout)[lane];
    float s = a.x * w.x + a.y * w.y + a.z * w.z + a.w * w.w;
    #pragma unroll
    for (int off = 16; off > 0; off >>= 1) s += __shfl_xor(s, off, 32);
    if (lane == 0) out[node] = s + bout[0];
}

// ---------------- launch ----------------
extern "C" void kernel_launch(void* const* d_in, const int* in_sizes, int n_in,
                              void* d_out, int out_size, void* d_ws, size_t ws_size,
                              hipStream_t stream) {
    const float* x    = (const float*)d_in[0];
    const int*   ei   = (const int*)  d_in[1];
    const int*   src  = ei;
    const int*   dst  = ei + EE;
    const float* Wl1  = (const float*)d_in[2];
    const float* bl1  = (const float*)d_in[3];
    const float* Wr1  = (const float*)d_in[4];
    const float* Wl2  = (const float*)d_in[5];
    const float* bl2  = (const float*)d_in[6];
    const float* Wr2  = (const float*)d_in[7];
    const float* Wl3  = (const float*)d_in[8];
    const float* bl3  = (const float*)d_in[9];
    const float* Wr3  = (const float*)d_in[10];
    const float* Wout = (const float*)d_in[11];
    const float* bout = (const float*)d_in[12];
    float* out = (float*)d_out;

    // workspace layout (floats): invdeg[NN] | agg[NN*HIDF] | hA[NN*HIDF] | hB[NN*HIDF]
    float* ws     = (float*)d_ws;
    float* invdeg = ws;
    float* agg    = invdeg + NN;
    float* hA     = agg + (size_t)NN * HIDF;
    float* hB     = hA  + (size_t)NN * HIDF;

    const dim3 gemmGrid((NN + 255) / 256, HIDF / 16);

    // degrees -> inverse degrees (shared by all layers)
    hipMemsetAsync(invdeg, 0, NN * sizeof(float), stream);
    deg_kernel<<<(EE + 255) / 256, 256, 0, stream>>>(dst, invdeg);
    invdeg_kernel<<<(NN + 255) / 256, 256, 0, stream>>>(invdeg);

    // ---- layer 1: x (N x 12) -> hA (N x 128) ----
    hipMemsetAsync(agg, 0, (size_t)NN * TOPF * sizeof(float), stream);
    {
        const int total = EE * (TOPF / 4);
        scatter_kernel<TOPF><<<(total + 255) / 256, 256, 0, stream>>>(x, src, dst, agg);
    }
    sage_gemm_kernel<TOPF, true><<<gemmGrid, 256, 0, stream>>>(
        agg, invdeg, x, Wl1, bl1, Wr1, hA);

    // ---- layer 2: hA -> hB ----
    hipMemsetAsync(agg, 0, (size_t)NN * HIDF * sizeof(float), stream);
    {
        const int total = EE * (HIDF / 4);
        scatter_kernel<HIDF><<<(total + 255) / 256, 256, 0, stream>>>(hA, src, dst, agg);
    }
    sage_gemm_kernel<HIDF, true><<<gemmGrid, 256, 0, stream>>>(
        agg, invdeg, hA, Wl2, bl2, Wr2, hB);

    // ---- layer 3: hB -> hA ----
    hipMemsetAsync(agg, 0, (size_t)NN * HIDF * sizeof(float), stream);
    {
        const int total = EE * (HIDF / 4);
        scatter_kernel<HIDF><<<(total + 255) / 256, 256, 0, stream>>>(hB, src, dst, agg);
    }
    sage_gemm_kernel<HIDF, true><<<gemmGrid, 256, 0, stream>>>(
        agg, invdeg, hB, Wl3, bl3, Wr3, hA);

    // ---- head: hA @ Wout.T + bout -> out (N x 1) ----
    head_kernel<<<(NN * 32 + 255) / 256, 256, 0, stream>>>(hA, Wout, bout, out);
}